// RouterCompoundFast_41558103556216
// MI455X (gfx1250) — compile-verified
//
#include <hip/hip_runtime.h>
#include <hip/hip_bf16.h>
#include <math.h>

typedef __attribute__((ext_vector_type(2))) float v2f;
typedef __attribute__((ext_vector_type(8))) float v8f;

#define BS      1024
#define HIDDEN  1024
#define NEXP    64
#define ROWS    32          // BIG*INNER rows per expert matrix
#define TOPK    4
#define NPAIR   (BS*TOPK)   // 4096 (token,slot) pairs
#define CAP     BS          // per-expert capacity: top-4 experts are distinct,
                            // so an expert gets at most one slot per token
#define MAXTILE (CAP/16)    // 64 row-tiles max per expert
#define INNER   8
#define BIG     4
#define MAXK    3
#define TOTAL   8

__device__ __forceinline__ int imin(int a, int b) { return a < b ? a : b; }

// ---------------------------------------------------------------------------
// Kernel 1: logits = x @ out_gate_weight^T  via V_WMMA_F32_16X16X4_F32
// grid = 64 (M tiles of 16), block = 128 (4 waves, one 16-expert N tile each)
// ---------------------------------------------------------------------------
__global__ void router_gemm(const float* __restrict__ x,
                            const float* __restrict__ gw,
                            float* __restrict__ logits) {
  const int wave  = threadIdx.x >> 5;
  const int lane  = threadIdx.x & 31;
  const int mt    = blockIdx.x;          // 0..63
  const int nt    = wave;                // 0..3
  const int lrow  = lane & 15;
  const int khalf = lane >> 4;           // 0: K={0,1}, 1: K={2,3}

  const float* arow = x  + (size_t)(mt * 16 + lrow) * HIDDEN;
  const float* brow = gw + (size_t)(nt * 16 + lrow) * HIDDEN;

  v8f acc = {};
  for (int k = 0; k < HIDDEN; k += 4) {
    v2f a = *(const v2f*)(arow + k + 2 * khalf);
    v2f b = *(const v2f*)(brow + k + 2 * khalf);
    acc = __builtin_amdgcn_wmma_f32_16x16x4_f32(
        /*neg_a=*/false, a, /*neg_b=*/false, b,
        /*c_mod=*/(short)0, acc, /*reuse_a=*/false, /*reuse_b=*/false);
  }

  const int n = nt * 16 + (lane & 15);
  #pragma unroll
  for (int r = 0; r < 8; ++r) {
    const int m = mt * 16 + r + 8 * khalf;
    logits[m * NEXP + n] = acc[r];
  }
}

// ---------------------------------------------------------------------------
// Kernel 2: per-token one-pass top-4 over logits; renormalized softmax weights
// (the full-softmax denominator cancels); scatter into per-expert lists.
// ---------------------------------------------------------------------------
__global__ void route_topk(const float* __restrict__ logits,
                           int* __restrict__ sel, float* __restrict__ rw,
                           int* __restrict__ counts, int* __restrict__ lists) {
  const int t = blockIdx.x * blockDim.x + threadIdx.x;
  if (t >= BS) return;
  const float* lr = logits + (size_t)t * NEXP;

  float v[TOPK]; int id[TOPK];
  #pragma unroll
  for (int i = 0; i < TOPK; ++i) { v[i] = -INFINITY; id[i] = -1; }

  for (int e = 0; e < NEXP; ++e) {
    const float l = lr[e];
    if (l > v[TOPK - 1]) {            // strict '>' => lower index wins ties (jax)
      int p = TOPK - 1;
      while (p > 0 && l > v[p - 1]) { v[p] = v[p - 1]; id[p] = id[p - 1]; --p; }
      v[p] = l; id[p] = e;
    }
  }

  const float m = v[0];
  float w[TOPK], sum = 0.f;
  #pragma unroll
  for (int i = 0; i < TOPK; ++i) { w[i] = __expf(v[i] - m); sum += w[i]; }
  const float inv = 1.f / sum;

  #pragma unroll
  for (int s = 0; s < TOPK; ++s) {
    const int pair = t * TOPK + s;
    sel[pair] = id[s];
    rw[pair]  = w[s] * inv;
    const int pos = atomicAdd(&counts[id[s]], 1);
    lists[id[s] * CAP + pos] = pair;
  }
}

// ---------------------------------------------------------------------------
// Kernel 3: expert-grouped dense GEMM, tile-parallel.
// grid = (64 experts, 64 row-tiles), block = 128 (4 waves).
// Wave roles: (w>>1)=matrix (0:gate,1:up), (w&1)=column half (rows 0-15/16-31).
// Gathers 16 token rows from the expert list, f32 WMMA K-loop, scatters the
// 16x16 result per (token,slot). Blocks past the expert's tile count exit.
// ---------------------------------------------------------------------------
__global__ void expert_gemm(const float* __restrict__ x,
                            const float* __restrict__ Wg,
                            const float* __restrict__ Wu,
                            const int* __restrict__ counts,
                            const int* __restrict__ lists,
                            float* __restrict__ gate_ws,
                            float* __restrict__ up_ws) {
  const int e    = blockIdx.x;
  const int tile = blockIdx.y;
  const int cnt  = counts[e];
  if (tile * 16 >= cnt) return;

  const int wave  = threadIdx.x >> 5;
  const int lane  = threadIdx.x & 31;
  const int mat   = wave >> 1;           // 0: gate, 1: up
  const int nh    = wave & 1;            // column half
  const int lrow  = lane & 15;
  const int khalf = lane >> 4;

  const float* W = (mat == 0 ? Wg : Wu) + (size_t)e * ROWS * HIDDEN;
  const int* lst = lists + e * CAP;
  float* outw    = (mat == 0 ? gate_ws : up_ws);

  const float* brow = W + (size_t)(nh * 16 + lrow) * HIDDEN;

  const int gidx  = imin(tile * 16 + lrow, cnt - 1);
  const int pairA = lst[gidx];
  const float* arow = x + (size_t)(pairA >> 2) * HIDDEN;   // token = pair/TOPK

  v8f acc = {};
  for (int k = 0; k < HIDDEN; k += 4) {
    v2f a = *(const v2f*)(arow + k + 2 * khalf);
    v2f b = *(const v2f*)(brow + k + 2 * khalf);
    acc = __builtin_amdgcn_wmma_f32_16x16x4_f32(
        false, a, false, b, (short)0, acc, false, false);
  }

  const int n = nh * 16 + (lane & 15);
  #pragma unroll
  for (int r = 0; r < 8; ++r) {
    const int mrow = tile * 16 + r + 8 * khalf;
    if (mrow < cnt) {
      const int pair = lst[mrow];
      outw[pair * ROWS + n] = acc[r];
    }
  }
}

// ---------------------------------------------------------------------------
// Kernel 4: silu/abs/mean-pool -> inner top-k (pattern-masked) -> the final
// top-8 over masked candidates reduces to descending sorts of the 8 valid
// ids and 8 valid weights. Output: [BS*TOTAL] ids (as float), then weights.
// ---------------------------------------------------------------------------
__global__ void finalize(const float* __restrict__ gate_ws,
                         const float* __restrict__ up_ws,
                         const int* __restrict__ sel,
                         const float* __restrict__ rw,
                         float* __restrict__ out) {
  const int t = blockIdx.x * blockDim.x + threadIdx.x;
  if (t >= BS) return;

  const int PAT[TOPK] = {3, 2, 2, 1};
  float ids[TOTAL], wts[TOTAL];
  int nc = 0;

  for (int s = 0; s < TOPK; ++s) {
    const int pair = t * TOPK + s;
    const float* g = gate_ws + (size_t)pair * ROWS;
    const float* u = up_ws   + (size_t)pair * ROWS;

    float inner[INNER];
    #pragma unroll
    for (int i = 0; i < INNER; ++i) {
      float acc = 0.f;
      #pragma unroll
      for (int j = 0; j < BIG; ++j) {
        const float gv = g[i * BIG + j];
        const float uv = u[i * BIG + j];
        const float silu = gv / (1.f + __expf(-gv));
        acc += fabsf(uv * silu);
      }
      inner[i] = acc * (1.f / BIG);
    }

    const int   e = sel[pair];
    const float w = rw[pair];

    bool used[INNER];
    #pragma unroll
    for (int i = 0; i < INNER; ++i) used[i] = false;

    for (int kk = 0; kk < PAT[s]; ++kk) {
      int best = 0; float bv = -INFINITY;
      #pragma unroll
      for (int i = 0; i < INNER; ++i)
        if (!used[i] && inner[i] > bv) { bv = inner[i]; best = i; }
      used[best] = true;
      ids[nc] = (float)(e * INNER + best);
      wts[nc] = w;
      ++nc;
    }
  }

  // descending insertion sorts (8 elements each)
  #pragma unroll
  for (int i = 1; i < TOTAL; ++i) {
    const float key = ids[i]; int p = i - 1;
    while (p >= 0 && ids[p] < key) { ids[p + 1] = ids[p]; --p; }
    ids[p + 1] = key;
  }
  #pragma unroll
  for (int i = 1; i < TOTAL; ++i) {
    const float key = wts[i]; int p = i - 1;
    while (p >= 0 && wts[p] < key) { wts[p + 1] = wts[p]; --p; }
    wts[p + 1] = key;
  }

  #pragma unroll
  for (int j = 0; j < TOTAL; ++j) {
    out[t * TOTAL + j]              = ids[j];
    out[BS * TOTAL + t * TOTAL + j] = wts[j];
  }
}

// ---------------------------------------------------------------------------
extern "C" void kernel_launch(void* const* d_in, const int* in_sizes, int n_in,
                              void* d_out, int out_size, void* d_ws, size_t ws_size,
                              hipStream_t stream) {
  const float* x  = (const float*)d_in[0];   // [BS, HIDDEN]
  const float* gw = (const float*)d_in[1];   // [NEXP, HIDDEN]
  const float* Wg = (const float*)d_in[2];   // [NEXP, ROWS, HIDDEN]
  const float* Wu = (const float*)d_in[3];   // [NEXP, ROWS, HIDDEN]
  float* out = (float*)d_out;

  // workspace layout (all 4-byte types)
  char* ws = (char*)d_ws;
  float* logits  = (float*)ws;                 ws += (size_t)BS * NEXP * 4;
  int*   sel     = (int*)ws;                   ws += (size_t)NPAIR * 4;
  float* rw      = (float*)ws;                 ws += (size_t)NPAIR * 4;
  int*   counts  = (int*)ws;                   ws += (size_t)NEXP * 4;
  int*   lists   = (int*)ws;                   ws += (size_t)NEXP * CAP * 4;
  float* gate_ws = (float*)ws;                 ws += (size_t)NPAIR * ROWS * 4;
  float* up_ws   = (float*)ws;                 ws += (size_t)NPAIR * ROWS * 4;

  hipMemsetAsync(counts, 0, NEXP * sizeof(int), stream);

  router_gemm<<<dim3(BS / 16), dim3(128), 0, stream>>>(x, gw, logits);
  route_topk<<<dim3((BS + 255) / 256), dim3(256), 0, stream>>>(logits, sel, rw,
                                                               counts, lists);
  expert_gemm<<<dim3(NEXP, MAXTILE), dim3(128), 0, stream>>>(x, Wg, Wu, counts,
                                                             lists, gate_ws, up_ws);
  finalize<<<dim3((BS + 255) / 256), dim3(256), 0, stream>>>(gate_ws, up_ws,
                                                             sel, rw, out);
}